// DrugGAT_56607668961829
// MI455X (gfx1250) — compile-verified
//
#include <hip/hip_runtime.h>
#include <math.h>

typedef float v2f __attribute__((ext_vector_type(2)));
typedef float v8f __attribute__((ext_vector_type(8)));

#define NEG_SLOPE 0.2f
#define BN_EPS 1e-5f
#define PS 72   // LDS panel row stride (64 data + 8 pad): 2*72 % 64 == 16 -> conflict-free halves

// ---------------- utility ----------------
__global__ void k_zero(float* __restrict__ p, long n) {
    long i = (long)blockIdx.x * blockDim.x + threadIdx.x;
    long stride = (long)gridDim.x * blockDim.x;
    for (; i < n; i += stride) p[i] = 0.f;
}

// ---------------- dual GEMM: XL = X*Wl, XR = X*Wr --------------
// X[M,K] row-major, W[K,C] row-major. Workgroup = 8 waves sharing one 64-col
// weight panel pair staged in LDS; each wave owns a 16-row M-tile.
// V_WMMA_F32_16X16X4_F32, ISA sec 7.12.2 fragment layouts.
template <int K>
__global__ void __launch_bounds__(256)
k_gemm_dual(const float* __restrict__ X,
            const float* __restrict__ Wl,
            const float* __restrict__ Wr,
            float* __restrict__ XL, float* __restrict__ XR,
            int M, int C) {
    __shared__ float lL[K * PS];
    __shared__ float lR[K * PS];

    int n0 = blockIdx.y * 64;

    // ---- cooperative panel stage: 256 threads, b128 per thread per row-group ----
    {
        int tid = threadIdx.y * 32 + threadIdx.x;   // 0..255
        int rr = tid >> 4;                          // row within 16-row pass
        int cc = (tid & 15) * 4;                    // 4-float column chunk
        for (int k0 = 0; k0 < K; k0 += 16) {
            int kr = k0 + rr;
            float4 vl = *(const float4*)&Wl[(long)kr * C + n0 + cc];
            float4 vr = *(const float4*)&Wr[(long)kr * C + n0 + cc];
            *(float4*)&lL[kr * PS + cc] = vl;
            *(float4*)&lR[kr * PS + cc] = vr;
        }
    }
    __syncthreads();

    int wave = blockIdx.x * blockDim.y + threadIdx.y;
    int m0 = wave * 16;
    if (m0 >= M) return;

    int lane = threadIdx.x;      // 0..31
    int hi   = lane >> 4;        // half-wave: K pair select / M+8 on store
    int nl   = lane & 15;
    int row  = m0 + nl;          // A: both halves hold rows M=0..15

    v8f aL[4] = {};
    v8f aR[4] = {};

    for (int k = 0; k < K; k += 4) {
        int kb = k + 2 * hi;     // lanes<16 -> K k,k+1 ; lanes>=16 -> K k+2,k+3
        v2f a;
        a.x = X[(long)row * K + kb];
        a.y = X[(long)row * K + kb + 1];
#pragma unroll
        for (int t = 0; t < 4; ++t) {
            int n = t * 16 + nl;
            v2f bl, br;
            bl.x = lL[kb * PS + n];
            bl.y = lL[(kb + 1) * PS + n];
            aL[t] = __builtin_amdgcn_wmma_f32_16x16x4_f32(
                false, a, false, bl, (short)0, aL[t], false, false);
            br.x = lR[kb * PS + n];
            br.y = lR[(kb + 1) * PS + n];
            aR[t] = __builtin_amdgcn_wmma_f32_16x16x4_f32(
                false, a, false, br, (short)0, aR[t], false, false);
        }
    }
    // C/D layout: VGPR i -> M = i (lanes 0-15) / M = 8+i (lanes 16-31), N = lane&15
#pragma unroll
    for (int t = 0; t < 4; ++t) {
        int n = n0 + t * 16 + nl;
#pragma unroll
        for (int i = 0; i < 8; ++i) {
            long r = m0 + i + 8 * hi;
            XL[r * C + n] = aL[t][i];
            XR[r * C + n] = aR[t][i];
        }
    }
}

// ---------------- edge pass 1: logits + segment max (warp/edge) --------------
template <int H, int C>
__global__ void k_edge_logits(const float* __restrict__ XL, const float* __restrict__ XR,
                              const float* __restrict__ att,
                              const int* __restrict__ srcA, const int* __restrict__ dstA,
                              float* __restrict__ logits, unsigned* __restrict__ menc,
                              int E, int N) {
    int lane = threadIdx.x & 31;
    long gid = (long)blockIdx.x * blockDim.x + threadIdx.x;
    int e = (int)(gid >> 5);
    int ET = E + N;
    if (e >= ET) return;
    int s = (e < E) ? srcA[e] : (e - E);   // self-loops appended
    int d = (e < E) ? dstA[e] : (e - E);
    const float* xl = XL + (long)s * (H * C);
    const float* xr = XR + (long)d * (H * C);
#pragma unroll
    for (int h = 0; h < H; ++h) {
        float p = 0.f;
#pragma unroll
        for (int c0 = 0; c0 < C; c0 += 32) {
            int c = c0 + lane;
            float v = xl[h * C + c] + xr[h * C + c];
            v = (v > 0.f) ? v : NEG_SLOPE * v;
            p += att[h * C + c] * v;
        }
#pragma unroll
        for (int off = 16; off > 0; off >>= 1) p += __shfl_xor(p, off, 32);
        if (lane == 0) {
            logits[(long)e * H + h] = p;
            unsigned u = __float_as_uint(p);           // monotone float->uint
            u = (u & 0x80000000u) ? ~u : (u | 0x80000000u);
            atomicMax(&menc[(long)d * H + h], u);
        }
    }
}

// ---------------- edge pass 2: exp + segment sum -----------------------------
template <int H>
__global__ void k_edge_exp(float* __restrict__ exbuf, const unsigned* __restrict__ menc,
                           float* __restrict__ denom, const int* __restrict__ dstA,
                           int E, int N) {
    long idx = (long)blockIdx.x * blockDim.x + threadIdx.x;
    long total = (long)(E + N) * H;
    if (idx >= total) return;
    int e = (int)(idx / H);
    int h = (int)(idx - (long)e * H);
    int d = (e < E) ? dstA[e] : (e - E);
    unsigned u = menc[(long)d * H + h];
    float m = __uint_as_float((u & 0x80000000u) ? (u ^ 0x80000000u) : ~u);
    float ex = expf(exbuf[idx] - m);
    exbuf[idx] = ex;
    atomicAdd(&denom[(long)d * H + h], ex);
}

// ---------------- edge pass 3: weighted scatter aggregate (warp/edge) --------
template <int H, int C>
__global__ void k_edge_aggregate(const float* __restrict__ XL,
                                 const float* __restrict__ exbuf,
                                 const float* __restrict__ denom,
                                 const int* __restrict__ srcA, const int* __restrict__ dstA,
                                 float* __restrict__ out, int E, int N) {
    int lane = threadIdx.x & 31;
    long gid = (long)blockIdx.x * blockDim.x + threadIdx.x;
    int e = (int)(gid >> 5);
    int ET = E + N;
    if (e >= ET) return;
    int s = (e < E) ? srcA[e] : (e - E);
    int d = (e < E) ? dstA[e] : (e - E);
    const float* xl = XL + (long)s * (H * C);
    float* o = out + (long)d * (H * C);
#pragma unroll
    for (int h = 0; h < H; ++h) {
        float alpha = exbuf[(long)e * H + h] / denom[(long)d * H + h];
#pragma unroll
        for (int c0 = 0; c0 < C; c0 += 32) {
            int c = c0 + lane;
            atomicAdd(&o[h * C + c], alpha * xl[h * C + c]);
        }
    }
}

// ---------------- batchnorm ----------------
__global__ void k_bn_stats(const float* __restrict__ h, float* __restrict__ sum,
                           float* __restrict__ sumsq, int Nn, int HC) {
    int c = threadIdx.x;          // blockDim.x == HC
    int r0 = blockIdx.x * 64;
    int r1 = r0 + 64; if (r1 > Nn) r1 = Nn;
    float s = 0.f, s2 = 0.f;
    for (int r = r0; r < r1; ++r) {
        float v = h[(long)r * HC + c];
        s += v; s2 += v * v;
    }
    atomicAdd(&sum[c], s);
    atomicAdd(&sumsq[c], s2);
}

__global__ void k_bn_finalize(const float* __restrict__ sum, const float* __restrict__ sumsq,
                              const float* __restrict__ gamma, const float* __restrict__ beta,
                              float* __restrict__ scale, float* __restrict__ shift, int Nn) {
    int c = threadIdx.x;
    float mean = sum[c] / (float)Nn;
    float var = sumsq[c] / (float)Nn - mean * mean;
    float sc = gamma[c] * rsqrtf(var + BN_EPS);
    scale[c] = sc;
    shift[c] = beta[c] - mean * sc;
}

__global__ void k_bn_apply_elu(float* __restrict__ h, const float* __restrict__ scale,
                               const float* __restrict__ shift, long total, int HC) {
    long i = (long)blockIdx.x * blockDim.x + threadIdx.x;
    if (i >= total) return;
    int c = (int)(i % HC);
    float v = h[i] * scale[c] + shift[c];
    h[i] = (v > 0.f) ? v : (expf(v) - 1.f);
}

__global__ void k_bias_add(float* __restrict__ out, const float* __restrict__ b,
                           long total, int C) {
    long i = (long)blockIdx.x * blockDim.x + threadIdx.x;
    if (i >= total) return;
    out[i] += b[(int)(i % C)];
}

// ---------------- driver ----------------
extern "C" void kernel_launch(void* const* d_in, const int* in_sizes, int n_in,
                              void* d_out, int out_size, void* d_ws, size_t ws_size,
                              hipStream_t stream) {
    const float* x    = (const float*)d_in[0];
    const int*   ei   = (const int*)d_in[1];
    const float* Wl1  = (const float*)d_in[2];
    const float* Wr1  = (const float*)d_in[3];
    const float* att1 = (const float*)d_in[4];
    // b1 (d_in[5]) cancels inside BatchNorm -> skipped
    const float* g1   = (const float*)d_in[6];
    const float* be1  = (const float*)d_in[7];
    const float* Wl2  = (const float*)d_in[8];
    const float* Wr2  = (const float*)d_in[9];
    const float* att2 = (const float*)d_in[10];
    // b2 (d_in[11]) cancels inside BatchNorm -> skipped
    const float* g2   = (const float*)d_in[12];
    const float* be2  = (const float*)d_in[13];
    const float* Wl3  = (const float*)d_in[14];
    const float* Wr3  = (const float*)d_in[15];
    const float* att3 = (const float*)d_in[16];
    const float* b3   = (const float*)d_in[17];

    const int IN_F = 128, HEADS = 4, OUT_F = 128, HC = 256;
    int N = in_sizes[0] / IN_F;
    int E = in_sizes[1] / 2;
    int ET = E + N;
    const int* srcA = ei;        // edge_index[0]
    const int* dstA = ei + E;    // edge_index[1]
    float* out = (float*)d_out;

    // workspace carve-up (floats)
    float* ws = (float*)d_ws;
    float* h1    = ws;                 ws += (long)N * HC;
    float* h2    = ws;                 ws += (long)N * HC;
    float* xl    = ws;                 ws += (long)N * HC;
    float* xr    = ws;                 ws += (long)N * HC;
    float* exbuf = ws;                 ws += (long)ET * HEADS;
    float* denom = ws;                 ws += (long)N * HEADS;
    unsigned* menc = (unsigned*)ws;    ws += (long)N * HEADS;
    float* bnsum = ws;                 ws += HC;
    float* bnsq  = ws;                 ws += HC;
    float* scale = ws;                 ws += HC;
    float* shift = ws;                 ws += HC;

    auto zero = [&](float* p, long n) {
        long b = (n + 255) / 256; if (b > 65535) b = 65535;
        k_zero<<<(int)b, 256, 0, stream>>>(p, n);
    };

    dim3 gblk(32, 8);                                  // 8 waves / workgroup
    int mwaves = (N + 15) / 16;
    int ggx = (mwaves + 7) / 8;
    int eblocks = (int)(((long)ET * 32 + 255) / 256);  // warp per edge
    long tHC = (long)N * HC;
    long tOF = (long)N * OUT_F;

    // ================= Layer 1 =================
    k_gemm_dual<128><<<dim3(ggx, HC / 64), gblk, 0, stream>>>(x, Wl1, Wr1, xl, xr, N, HC);
    zero(h1, tHC);
    zero(denom, (long)N * HEADS);
    zero((float*)menc, (long)N * HEADS);
    k_edge_logits<4, 64><<<eblocks, 256, 0, stream>>>(xl, xr, att1, srcA, dstA, exbuf, menc, E, N);
    {
        long t = (long)ET * HEADS;
        k_edge_exp<4><<<(int)((t + 255) / 256), 256, 0, stream>>>(exbuf, menc, denom, dstA, E, N);
    }
    k_edge_aggregate<4, 64><<<eblocks, 256, 0, stream>>>(xl, exbuf, denom, srcA, dstA, h1, E, N);
    zero(bnsum, HC); zero(bnsq, HC);
    k_bn_stats<<<(N + 63) / 64, HC, 0, stream>>>(h1, bnsum, bnsq, N, HC);
    k_bn_finalize<<<1, HC, 0, stream>>>(bnsum, bnsq, g1, be1, scale, shift, N);
    k_bn_apply_elu<<<(int)((tHC + 255) / 256), 256, 0, stream>>>(h1, scale, shift, tHC, HC);

    // ================= Layer 2 =================
    k_gemm_dual<256><<<dim3(ggx, HC / 64), gblk, 0, stream>>>(h1, Wl2, Wr2, xl, xr, N, HC);
    zero(h2, tHC);
    zero(denom, (long)N * HEADS);
    zero((float*)menc, (long)N * HEADS);
    k_edge_logits<4, 64><<<eblocks, 256, 0, stream>>>(xl, xr, att2, srcA, dstA, exbuf, menc, E, N);
    {
        long t = (long)ET * HEADS;
        k_edge_exp<4><<<(int)((t + 255) / 256), 256, 0, stream>>>(exbuf, menc, denom, dstA, E, N);
    }
    k_edge_aggregate<4, 64><<<eblocks, 256, 0, stream>>>(xl, exbuf, denom, srcA, dstA, h2, E, N);
    zero(bnsum, HC); zero(bnsq, HC);
    k_bn_stats<<<(N + 63) / 64, HC, 0, stream>>>(h2, bnsum, bnsq, N, HC);
    k_bn_finalize<<<1, HC, 0, stream>>>(bnsum, bnsq, g2, be2, scale, shift, N);
    k_bn_apply_elu<<<(int)((tHC + 255) / 256), 256, 0, stream>>>(h2, scale, shift, tHC, HC);

    // ================= Layer 3 (1 head, 128 ch, mean==identity) =================
    k_gemm_dual<256><<<dim3(ggx, OUT_F / 64), gblk, 0, stream>>>(h2, Wl3, Wr3, xl, xr, N, OUT_F);
    zero(out, tOF);
    zero(denom, (long)N);
    zero((float*)menc, (long)N);
    k_edge_logits<1, 128><<<eblocks, 256, 0, stream>>>(xl, xr, att3, srcA, dstA, exbuf, menc, E, N);
    {
        long t = (long)ET;
        k_edge_exp<1><<<(int)((t + 255) / 256), 256, 0, stream>>>(exbuf, menc, denom, dstA, E, N);
    }
    k_edge_aggregate<1, 128><<<eblocks, 256, 0, stream>>>(xl, exbuf, denom, srcA, dstA, out, E, N);
    k_bias_add<<<(int)((tOF + 255) / 256), 256, 0, stream>>>(out, b3, tOF, OUT_F);
}